// TSClusteringLayer_17609365913690
// MI455X (gfx1250) — compile-verified
//
#include <hip/hip_runtime.h>
#include <hip/hip_bf16.h>
#include <math.h>

typedef __attribute__((ext_vector_type(2))) float v2f;
typedef __attribute__((ext_vector_type(4))) float v4f;
typedef __attribute__((ext_vector_type(8))) float v8f;

#define B_N 2048
#define T_N 128
#define F_N 16
#define K_N 32
#define FG  4     // features per wave (one b128 slice of the contiguous F axis)

// ---------------------------------------------------------------------------
// Kernel 1: dist[b,k] = sum_f sqrt( sum_t (x[b,t,f]-c[k,t,f])^2 )
// Block = 128 threads = 4 waves; block owns one (16-row batch, 16-col cluster)
// tile; wave w owns features [4w, 4w+4).  Per feature: cross term via chained
// V_WMMA_F32_16X16X4_F32 over t (A pre-scaled by -2), then one augmented
// rank-2 WMMA folds ||x||^2 + ||c||^2 in, so the accumulator holds d^2.
// Partial per-f-group distance tiles are summed deterministically via LDS.
// ---------------------------------------------------------------------------
__global__ __launch_bounds__(128)
void ts_dist_kernel(const float* __restrict__ x, const float* __restrict__ c,
                    float* __restrict__ dist) {
  __shared__ float ptile[4 * 256];        // 4 partial 16x16 tiles

  const int mtile = blockIdx.x >> 1;      // 0..127
  const int ntile = blockIdx.x & 1;       // 0..1
  const int m0 = mtile * 16;
  const int n0 = ntile * 16;

  const int tid   = threadIdx.x;
  const int wave  = tid >> 5;             // f-group id 0..3
  const int lane  = tid & 31;
  const int lrow  = lane & 15;            // matrix row this lane feeds
  const int lhalf = lane >> 4;            // 0: K=0,1   1: K=2,3 (A/B layout)

  // Each lane streams one batch row (A) and one cluster row (B), 16B/b128 per t.
  const float* __restrict__ xrow =
      x + (size_t)(m0 + lrow) * (T_N * F_N) + wave * FG;
  const float* __restrict__ crow =
      c + (size_t)(n0 + lrow) * (T_N * F_N) + wave * FG;

  v8f   acc[FG];
  float sx2[FG], sc2[FG];
  #pragma unroll
  for (int i = 0; i < FG; ++i) {
    acc[i] = v8f{0.f,0.f,0.f,0.f,0.f,0.f,0.f,0.f};
    sx2[i] = 0.f;
    sc2[i] = 0.f;
  }

  #pragma unroll 2
  for (int tc = 0; tc < T_N; tc += 4) {
    const int tA = tc + 2 * lhalf;        // lanes 0-15: t,t+1   lanes 16-31: t+2,t+3
    const v4f a0 = *(const v4f*)(xrow +  tA      * F_N);   // global_load_b128
    const v4f a1 = *(const v4f*)(xrow + (tA + 1) * F_N);
    const v4f b0 = *(const v4f*)(crow +  tA      * F_N);
    const v4f b1 = *(const v4f*)(crow + (tA + 1) * F_N);

    #pragma unroll
    for (int i = 0; i < FG; ++i) {
      sx2[i] = fmaf(a0[i], a0[i], fmaf(a1[i], a1[i], sx2[i]));
      sc2[i] = fmaf(b0[i], b0[i], fmaf(b1[i], b1[i], sc2[i]));

      v2f av; av.x = -2.0f * a0[i]; av.y = -2.0f * a1[i];  // fold the -2 into A
      v2f bv; bv.x = b0[i];         bv.y = b1[i];

      acc[i] = __builtin_amdgcn_wmma_f32_16x16x4_f32(
          /*neg_a=*/false, av, /*neg_b=*/false, bv,
          /*c_mod=*/(short)0, acc[i], /*reuse_a=*/false, /*reuse_b=*/false);
    }
  }

  // Per feature: fold norms with an augmented rank-2 WMMA, then sqrt-accumulate.
  v8f dacc = v8f{0.f,0.f,0.f,0.f,0.f,0.f,0.f,0.f};
  #pragma unroll
  for (int i = 0; i < FG; ++i) {
    const float sx = sx2[i] + __shfl_xor(sx2[i], 16, 32);  // combine t%4 halves
    const float sc = sc2[i] + __shfl_xor(sc2[i], 16, 32);

    v2f av, bv;                    // A'=[||x||^2, 1, 0, 0]  B'=[1, ||c||^2, 0, 0]^T
    av.x = lhalf ? 0.f : sx;
    av.y = lhalf ? 0.f : 1.f;
    bv.x = lhalf ? 0.f : 1.f;
    bv.y = lhalf ? 0.f : sc;
    acc[i] = __builtin_amdgcn_wmma_f32_16x16x4_f32(
        false, av, false, bv, (short)0, acc[i], false, false);

    #pragma unroll
    for (int j = 0; j < 8; ++j) {
      // v_max_num_f32 + v_sqrt_f32 (hardware sqrt, ~1 ulp) instead of the
      // IEEE software-expansion sequence.
      const float d2 = fmaxf(acc[i][j], 0.0f);
      dacc[j] += __builtin_amdgcn_sqrtf(d2);
    }
  }

  // Stash this wave's partial tile: element e = m_in_tile*16 + n_in_tile.
  #pragma unroll
  for (int j = 0; j < 8; ++j)
    ptile[wave * 256 + (j + 8 * lhalf) * 16 + lrow] = dacc[j];

  __syncthreads();

  // Deterministic fixed-order combine of the 4 f-group partials.
  for (int e = tid; e < 256; e += 128) {
    const float v = ((ptile[e] + ptile[256 + e]) + ptile[512 + e]) + ptile[768 + e];
    const int m = e >> 4;
    const int n = e & 15;
    dist[(size_t)(m0 + m) * K_N + (n0 + n)] = v;
  }
}

// ---------------------------------------------------------------------------
// Kernel 2 (in place on d_out): q = 1/(1+dist^2); q /= sum_k q.
// One wave per batch row; K=32 == wave32, row-sum via 5 xor-shuffles.
// ---------------------------------------------------------------------------
__global__ __launch_bounds__(256)
void ts_norm_kernel(float* __restrict__ q) {
  const int wave = (int)((blockIdx.x * blockDim.x + threadIdx.x) >> 5);  // row
  const int lane = threadIdx.x & 31;                                     // k

  const float d  = q[(size_t)wave * K_N + lane];
  const float qi = 1.0f / (1.0f + d * d);     // ALPHA = 1 -> exponent is 1

  float s = qi;
  s += __shfl_xor(s, 1, 32);
  s += __shfl_xor(s, 2, 32);
  s += __shfl_xor(s, 4, 32);
  s += __shfl_xor(s, 8, 32);
  s += __shfl_xor(s, 16, 32);

  q[(size_t)wave * K_N + lane] = qi / s;
}

// ---------------------------------------------------------------------------
extern "C" void kernel_launch(void* const* d_in, const int* in_sizes, int n_in,
                              void* d_out, int out_size, void* d_ws, size_t ws_size,
                              hipStream_t stream) {
  (void)in_sizes; (void)n_in; (void)out_size; (void)d_ws; (void)ws_size;
  const float* x = (const float*)d_in[0];   // inputs   (2048,128,16) f32
  const float* c = (const float*)d_in[1];   // clusters (32,128,16)  f32
  float* out = (float*)d_out;               // (2048,32) f32

  // 128 batch tiles x 2 cluster tiles, 4 f-group waves each -> 1024 waves
  ts_dist_kernel<<<dim3(256), dim3(128), 0, stream>>>(x, c, out);
  // 2048 rows, one wave each: 256 blocks x 256 threads
  ts_norm_kernel<<<dim3(256), dim3(256), 0, stream>>>(out);
}